// SparseEncoder_29850022708104
// MI455X (gfx1250) — compile-verified
//
#include <hip/hip_runtime.h>

typedef __attribute__((ext_vector_type(2))) float v2f;
typedef __attribute__((ext_vector_type(8))) float v8f;

#define BB   4
#define NN   16384
#define CC   64
#define MM   4096

typedef unsigned long long u64;
typedef unsigned int u32;

// Keep the 3 smallest u64 keys (t0 <= t1 <= t2). Lowers to straight-line
// v_min_u64/v_max_u64 on gfx1250 — no branches.
__device__ __forceinline__ void insert_key(u64 k, u64& t0, u64& t1, u64& t2) {
  u64 lo0 = (k  < t0) ? k  : t0;
  u64 hi0 = (k  < t0) ? t0 : k;
  u64 lo1 = (hi0 < t1) ? hi0 : t1;
  u64 hi1 = (hi0 < t1) ? t1  : hi0;
  u64 lo2 = (hi1 < t2) ? hi1 : t2;
  t0 = lo0; t1 = lo1; t2 = lo2;
}

__device__ __forceinline__ u64 shfl_xor_u64(u64 v, int m) {
  u32 lo = __shfl_xor((u32)v, m, 32);
  u32 hi = __shfl_xor((u32)(v >> 32), m, 32);
  return ((u64)hi << 32) | (u64)lo;
}

// key = (bits(clamp(d,0,inf)) << 32) | n : orders by d2, ties by lower index.
// v_med3_f32(d, 0, +inf) == max(d, 0) for finite d — single instruction,
// avoids the 2-op OCML fmaxf canonicalization.
__device__ __forceinline__ u64 make_key(float d, int n) {
  float dc = __builtin_amdgcn_fmed3f(d, 0.0f, __builtin_inff());
  return ((u64)__float_as_uint(dc) << 32) | (u32)n;
}

#define KEY_INIT 0x7F800000FFFFFFFFull  // hi = +inf bits

__global__ __launch_bounds__(256) void knn_interp_wmma(
    const float* __restrict__ x,       // [B, N, C]
    const float* __restrict__ coords,  // [B, N, 2]
    float* __restrict__ out) {         // [B, M, C]
  __shared__ int s_idx[8][16][3];      // per-wave slots

  const int lane = threadIdx.x & 31;
  const int wib  = threadIdx.x >> 5;                 // wave in block (0..7)
  const int wave = blockIdx.x * 8 + wib;             // 0..1023
  const int b      = wave >> 8;                      // batch (0..3)
  const int m_base = (wave & 255) << 4;              // 16 grid rows per wave

  const float* cb = coords + (long)b * NN * 2;

  // ---- A operand: row m = [-2*gy, -2*gx, |g|^2, 1] ---------------------
  // ISA 16x4 f32 A layout: lanes 0-15 hold (K0,K1) of row=lane,
  //                        lanes 16-31 hold (K2,K3) of row=lane-16.
  const int rowA = lane & 15;
  const int gm_a = m_base + rowA;
  const float gy = (float)(gm_a >> 6) * (1.0f / 63.0f);
  const float gx = (float)(gm_a & 63) * (1.0f / 63.0f);
  const float g2 = gy * gy + gx * gx;
  v2f A;
  A[0] = (lane < 16) ? (-2.0f * gy) : g2;
  A[1] = (lane < 16) ? (-2.0f * gx) : 1.0f;

  // ---- top-3 keys per lane for its 8 D-rows ----------------------------
  u64 k0[8], k1[8], k2[8];
#pragma unroll
  for (int r = 0; r < 8; ++r) { k0[r] = k1[r] = k2[r] = KEY_INIT; }

  const int col = lane & 15;

  // ---- main loop: 4 N-tiles (64 points) per iteration, pipelined -------
  float2 c[4];
#pragma unroll
  for (int t = 0; t < 4; ++t)
    c[t] = *(const float2*)(cb + (long)(t * 16 + col) * 2);

  for (int nb = 0; nb < NN; nb += 64) {
    // prefetch next super-tile (wrapped address; final prefetch unused)
    const int nb2 = (nb + 64) & (NN - 1);
    float2 cn[4];
#pragma unroll
    for (int t = 0; t < 4; ++t)
      cn[t] = *(const float2*)(cb + (long)(nb2 + t * 16 + col) * 2);

    v8f d[4];
#pragma unroll
    for (int t = 0; t < 4; ++t) {
      const float csq = c[t].x * c[t].x + c[t].y * c[t].y;
      // B col n = [c0, c1, 1, |c|^2]; lanes 0-15 (K0,K1), 16-31 (K2,K3)
      v2f Bv;
      Bv[0] = (lane < 16) ? c[t].x : 1.0f;
      Bv[1] = (lane < 16) ? c[t].y : csq;
      v8f z = {};
      d[t] = __builtin_amdgcn_wmma_f32_16x16x4_f32(
          false, A, false, Bv, (short)0, z, false, false);
    }

    // per-tile ballot: slow path pays only 8 inserts per triggering tile
#pragma unroll
    for (int t = 0; t < 4; ++t) {
      bool need = false;
#pragma unroll
      for (int r = 0; r < 8; ++r)
        need |= (d[t][r] < __uint_as_float((u32)(k2[r] >> 32)));
      if (__ballot(need) != 0ull) {
        const int n = nb + t * 16 + col;
#pragma unroll
        for (int r = 0; r < 8; ++r)
          insert_key(make_key(d[t][r], n), k0[r], k1[r], k2[r]);
      }
    }

#pragma unroll
    for (int t = 0; t < 4; ++t) c[t] = cn[t];
  }

  // ---- butterfly merge over the 16 columns (stays within each half) ----
#pragma unroll
  for (int m = 1; m <= 8; m <<= 1) {
#pragma unroll
    for (int r = 0; r < 8; ++r) {
      u64 b0 = shfl_xor_u64(k0[r], m);
      u64 b1 = shfl_xor_u64(k1[r], m);
      u64 b2 = shfl_xor_u64(k2[r], m);
      insert_key(b0, k0[r], k1[r], k2[r]);
      insert_key(b1, k0[r], k1[r], k2[r]);
      insert_key(b2, k0[r], k1[r], k2[r]);
    }
  }

  // D layout: VGPR r = row r (lanes 0-15) or row r+8 (lanes 16-31)
  if (lane == 0) {
#pragma unroll
    for (int r = 0; r < 8; ++r) {
      s_idx[wib][r][0] = (int)(u32)k0[r];
      s_idx[wib][r][1] = (int)(u32)k1[r];
      s_idx[wib][r][2] = (int)(u32)k2[r];
    }
  } else if (lane == 16) {
#pragma unroll
    for (int r = 0; r < 8; ++r) {
      s_idx[wib][8 + r][0] = (int)(u32)k0[r];
      s_idx[wib][8 + r][1] = (int)(u32)k1[r];
      s_idx[wib][8 + r][2] = (int)(u32)k2[r];
    }
  }
  __syncthreads();

  // ---- phase 2: gather + exact-weight blend (2 lanes per row) ----------
  const int row = lane >> 1;            // 0..15
  const int h   = lane & 1;             // channel half
  const int gm  = m_base + row;
  const float ggy = (float)(gm >> 6) * (1.0f / 63.0f);
  const float ggx = (float)(gm & 63) * (1.0f / 63.0f);
  const float* xb = x + (long)b * NN * CC;

  float facc[32];
#pragma unroll
  for (int q = 0; q < 32; ++q) facc[q] = 0.0f;
  float wsum = 0.0f;

#pragma unroll
  for (int k = 0; k < 3; ++k) {
    const int n = s_idx[wib][row][k];
    const float2 c2 = *(const float2*)(cb + (long)n * 2);
    const float dy = c2.x - ggy, dx = c2.y - ggx;
    const float sq = dy * dy + dx * dx;
    const float w  = 1.0f / fmaxf(sq, 1e-16f);
    wsum += w;
    const float4* xp = (const float4*)(xb + (long)n * CC + h * 32);
#pragma unroll
    for (int q = 0; q < 8; ++q) {
      float4 v = xp[q];
      facc[q * 4 + 0] += w * v.x;
      facc[q * 4 + 1] += w * v.y;
      facc[q * 4 + 2] += w * v.z;
      facc[q * 4 + 3] += w * v.w;
    }
  }

  const float inv = 1.0f / wsum;
  float4* op = (float4*)(out + ((long)b * MM + gm) * CC + h * 32);
#pragma unroll
  for (int q = 0; q < 8; ++q) {
    float4 v;
    v.x = facc[q * 4 + 0] * inv;
    v.y = facc[q * 4 + 1] * inv;
    v.z = facc[q * 4 + 2] * inv;
    v.w = facc[q * 4 + 3] * inv;
    op[q] = v;
  }
}

extern "C" void kernel_launch(void* const* d_in, const int* in_sizes, int n_in,
                              void* d_out, int out_size, void* d_ws, size_t ws_size,
                              hipStream_t stream) {
  const float* x      = (const float*)d_in[0];
  const float* coords = (const float*)d_in[1];
  float* out          = (float*)d_out;
  // 1024 waves total: one wave per (batch, 16-grid-point tile)
  hipLaunchKernelGGL(knn_interp_wmma, dim3(128), dim3(256), 0, stream,
                     x, coords, out);
}